// GNN_1769526526179
// MI455X (gfx1250) — compile-verified
//
#include <hip/hip_runtime.h>

#define N_NODES   50000
#define N_EDGES   800000
#define HIDDEN    64
#define N_GRAPHS  64
#define IN_DIM    3
#define N_TILES   (N_NODES / 16)   // 3125, exact

typedef float v2f __attribute__((ext_vector_type(2)));
typedef float v8f __attribute__((ext_vector_type(8)));

// ---------------------------------------------------------------- utilities
__global__ void zero_f32(float* __restrict__ p, int n) {
    int i = blockIdx.x * blockDim.x + threadIdx.x;
    if (i < n) p[i] = 0.0f;
}

// ------------------------------------------------- layer 1: 3-dim aggregate
__global__ void scatter_dim3(const float* __restrict__ x,
                             const int* __restrict__ src,
                             const int* __restrict__ dst,
                             float* __restrict__ agg3) {
    int e = blockIdx.x * blockDim.x + threadIdx.x;
    if (e >= N_EDGES) return;
    int s = src[e], d = dst[e];
#pragma unroll
    for (int k = 0; k < IN_DIM; ++k)
        atomicAdd(&agg3[d * IN_DIM + k], x[s * IN_DIM + k]);
}

__global__ void layer1_node(const float* __restrict__ x,
                            const float* __restrict__ agg3,
                            const float* __restrict__ Wrel,
                            const float* __restrict__ Wroot,
                            const float* __restrict__ bias,
                            float* __restrict__ h) {
    int idx = blockIdx.x * blockDim.x + threadIdx.x;
    if (idx >= N_NODES * HIDDEN) return;
    int node = idx >> 6, o = idx & 63;
    float s = bias[o];
#pragma unroll
    for (int k = 0; k < IN_DIM; ++k) {
        s += agg3[node * IN_DIM + k] * Wrel[k * HIDDEN + o];
        s += x[node * IN_DIM + k] * Wroot[k * HIDDEN + o];
    }
    h[idx] = fmaxf(s, 0.0f);
}

// ----------------------------------------- layers 2/3: 64-dim edge scatter
__global__ void scatter_dim64(const float* __restrict__ h,
                              const int* __restrict__ src,
                              const int* __restrict__ dst,
                              float* __restrict__ agg) {
    int idx = blockIdx.x * blockDim.x + threadIdx.x;
    if (idx >= N_EDGES * 16) return;
    int e = idx >> 4;
    int c = (idx & 15) << 2;          // feature chunk of 4
    int s = src[e], d = dst[e];
    const float4 v = *(const float4*)(h + (size_t)s * HIDDEN + c);
    float* a = agg + (size_t)d * HIDDEN + c;
    atomicAdd(a + 0, v.x);
    atomicAdd(a + 1, v.y);
    atomicAdd(a + 2, v.z);
    atomicAdd(a + 3, v.w);
}

// --------------------------------------------- WMMA fp32 GraphConv GEMM
// One wave32 per 16-node tile: out[16x64] = agg[16x64]@Wrel + h[16x64]@Wroot + b
// A layout (f32 16x4, ISA 7.12.2): lane<16 holds K=0(v0),1(v1); lane>=16 K=2,3.
// B layout (f32 4x16): lane<16 -> N=lane, K=0(v0),1(v1); lane>=16 -> K=2,3.
// C/D layout: VGPR r, lanes 0-15 -> M=r, N=lane; lanes 16-31 -> M=r+8, N=lane-16.
__global__ void __launch_bounds__(128)
graphconv_wmma(const float* __restrict__ agg,
               const float* __restrict__ hin,
               const float* __restrict__ Wrel,
               const float* __restrict__ Wroot,
               const float* __restrict__ bias,
               float* __restrict__ hout,
               int do_relu) {
    const int wave = threadIdx.x >> 5;
    const int tile = blockIdx.x * 4 + wave;
    if (tile >= N_TILES) return;                 // whole-wave exit, EXEC stays full
    const int lane  = threadIdx.x & 31;
    const int m     = lane & 15;
    const int kk    = (lane >> 4) << 1;          // 0 or 2
    const int node0 = tile * 16;

    const float* hrow = hin + (size_t)(node0 + m) * HIDDEN;
    const float* grow = agg + (size_t)(node0 + m) * HIDDEN;

    // Preload ALL A fragments (both matrices) so the in-place store is safe
    v2f aG[16], aH[16];
#pragma unroll
    for (int ks = 0; ks < 16; ++ks) {
        int k0 = ks * 4 + kk;
        aG[ks].x = grow[k0]; aG[ks].y = grow[k0 + 1];   // pair -> b64 load
        aH[ks].x = hrow[k0]; aH[ks].y = hrow[k0 + 1];
    }

    const int n     = lane & 15;
    const int rbase = (lane >> 4) * 8;
#pragma unroll
    for (int t = 0; t < 4; ++t) {                // 4 column tiles of 16
        float bv = bias[t * 16 + n];
        v8f acc;
#pragma unroll
        for (int r = 0; r < 8; ++r) acc[r] = bv; // bias folded into C
#pragma unroll
        for (int ks = 0; ks < 16; ++ks) {        // K = 64 in steps of 4
            int krow = ks * 4 + kk;
            v2f bG, bH;
            const float* wr = Wrel  + (size_t)krow * HIDDEN + t * 16 + n;
            const float* wo = Wroot + (size_t)krow * HIDDEN + t * 16 + n;
            bG.x = wr[0]; bG.y = wr[HIDDEN];
            bH.x = wo[0]; bH.y = wo[HIDDEN];
            acc = __builtin_amdgcn_wmma_f32_16x16x4_f32(
                false, aG[ks], false, bG, (short)0, acc, false, false);
            acc = __builtin_amdgcn_wmma_f32_16x16x4_f32(
                false, aH[ks], false, bH, (short)0, acc, false, false);
        }
        float* ocol = hout + (size_t)node0 * HIDDEN + t * 16 + n;
#pragma unroll
        for (int r = 0; r < 8; ++r) {
            float v = acc[r];
            if (do_relu) v = fmaxf(v, 0.0f);
            ocol[(size_t)(rbase + r) * HIDDEN] = v;
        }
    }
}

// ----------------------------------------------------------- mean pooling
__global__ void pool_scatter(const float* __restrict__ h,
                             const int* __restrict__ batch,
                             float* __restrict__ sums,
                             float* __restrict__ counts) {
    int idx = blockIdx.x * blockDim.x + threadIdx.x;
    if (idx >= N_NODES * HIDDEN) return;
    int i = idx >> 6, f = idx & 63;
    int g = batch[i];
    atomicAdd(&sums[g * HIDDEN + f], h[idx]);
    if (f == 0) atomicAdd(&counts[g], 1.0f);
}

__global__ void heads(const float* __restrict__ sums,
                      const float* __restrict__ counts,
                      const float* __restrict__ lin1w, const float* __restrict__ lin1b,
                      const float* __restrict__ lin2w, const float* __restrict__ lin2b,
                      float* __restrict__ out) {
    int g = threadIdx.x;
    if (g >= N_GRAPHS) return;
    float c = fmaxf(counts[g], 1.0f);
    float d1 = 0.0f, d2 = 0.0f;
#pragma unroll
    for (int f = 0; f < HIDDEN; ++f) {
        float p = sums[g * HIDDEN + f] / c;
        d1 += p * lin1w[f];
        d2 += p * lin2w[f];
    }
    out[g]            = d1 + lin1b[0];   // x1
    out[N_GRAPHS + g] = d2 + lin2b[0];   // x2
}

// ---------------------------------------------------------------- launcher
extern "C" void kernel_launch(void* const* d_in, const int* in_sizes, int n_in,
                              void* d_out, int out_size, void* d_ws, size_t ws_size,
                              hipStream_t stream) {
    const float* x      = (const float*)d_in[0];
    const int*   ei     = (const int*)d_in[1];
    const int*   batch  = (const int*)d_in[2];
    const float* W1rel  = (const float*)d_in[3];
    const float* W1root = (const float*)d_in[4];
    const float* b1     = (const float*)d_in[5];
    const float* W2rel  = (const float*)d_in[6];
    const float* W2root = (const float*)d_in[7];
    const float* b2     = (const float*)d_in[8];
    const float* W3rel  = (const float*)d_in[9];
    const float* W3root = (const float*)d_in[10];
    const float* b3     = (const float*)d_in[11];
    const float* lin1w  = (const float*)d_in[12];
    const float* lin1b  = (const float*)d_in[13];
    const float* lin2w  = (const float*)d_in[14];
    const float* lin2b  = (const float*)d_in[15];
    float* out = (float*)d_out;

    float* ws     = (float*)d_ws;
    float* agg    = ws;                                   // 50000*64
    float* h      = ws + (size_t)N_NODES * HIDDEN;        // 50000*64
    float* sums   = h  + (size_t)N_NODES * HIDDEN;        // 64*64
    float* counts = sums + N_GRAPHS * HIDDEN;             // 64 (contiguous w/ sums)

    const int* src = ei;                                  // edge_index[0, :]
    const int* dst = ei + N_EDGES;                        // edge_index[1, :]

    const int B = 256;
    // ---- layer 1 (aggregate 3-dim input) ----
    zero_f32<<<(N_NODES * IN_DIM + B - 1) / B, B, 0, stream>>>(agg, N_NODES * IN_DIM);
    scatter_dim3<<<(N_EDGES + B - 1) / B, B, 0, stream>>>(x, src, dst, agg);
    layer1_node<<<(N_NODES * HIDDEN + B - 1) / B, B, 0, stream>>>(x, agg, W1rel, W1root, b1, h);

    // ---- layer 2 ----
    zero_f32<<<(N_NODES * HIDDEN + B - 1) / B, B, 0, stream>>>(agg, N_NODES * HIDDEN);
    scatter_dim64<<<(N_EDGES * 16 + B - 1) / B, B, 0, stream>>>(h, src, dst, agg);
    graphconv_wmma<<<(N_TILES + 3) / 4, 128, 0, stream>>>(agg, h, W2rel, W2root, b2, h, 1);

    // ---- layer 3 ----
    zero_f32<<<(N_NODES * HIDDEN + B - 1) / B, B, 0, stream>>>(agg, N_NODES * HIDDEN);
    scatter_dim64<<<(N_EDGES * 16 + B - 1) / B, B, 0, stream>>>(h, src, dst, agg);
    graphconv_wmma<<<(N_TILES + 3) / 4, 128, 0, stream>>>(agg, h, W3rel, W3root, b3, h, 0);

    // ---- global mean pool + heads ----
    zero_f32<<<(N_GRAPHS * HIDDEN + N_GRAPHS + B - 1) / B, B, 0, stream>>>(sums,
                                                     N_GRAPHS * HIDDEN + N_GRAPHS);
    pool_scatter<<<(N_NODES * HIDDEN + B - 1) / B, B, 0, stream>>>(h, batch, sums, counts);
    heads<<<1, 64, 0, stream>>>(sums, counts, lin1w, lin1b, lin2w, lin2b, out);
}